// GINNet_Cora_34832184770974
// MI455X (gfx1250) — compile-verified
//
#include <hip/hip_runtime.h>
#include <hip/hip_bf16.h>

// ---------------------------------------------------------------------------
// GIN on 100K-node graph, MI455X (gfx1250, wave32).
// Memory-bound (scatter-add dominates: ~0.9 GB traffic -> ~40us floor at
// 23.3 TB/s). GEMMs (64->16->16->16->7) are tiny; done with fp32 WMMA
// (V_WMMA_F32_16X16X4_F32), one 16-node tile per wave, LDS round-trip to
// re-layout C (row-striped-across-lanes) into A (row-per-lane) between layers.
// Tail tiles handled by CLAMPING the load row (always-valid address, no
// divergence) and predicating only the final stores.
// ---------------------------------------------------------------------------

typedef __attribute__((ext_vector_type(2))) float v2f;
typedef __attribute__((ext_vector_type(8))) float v8f;

__device__ __forceinline__ v8f wmma_f32_k4(v2f a, v2f b, v8f c) {
    // D = A(16x4, f32) * B(4x16, f32) + C(16x16, f32)
    return __builtin_amdgcn_wmma_f32_16x16x4_f32(
        /*neg_a=*/false, a, /*neg_b=*/false, b,
        /*c_mod=*/(short)0, c, /*reuse_a=*/false, /*reuse_b=*/false);
}

// ---------------------------------------------------------------------------
// Edge scatter: agg[dst] += feat[src], feature dim 64. 16 threads per edge,
// each moves a float4 via 4 fp32 global atomics (resolve in L2).
// ---------------------------------------------------------------------------
__global__ void __launch_bounds__(256) gin_scatter64(
    const float* __restrict__ feat, const int* __restrict__ ei,
    float* __restrict__ agg, int nEdges)
{
    long long t = (long long)blockIdx.x * 256 + threadIdx.x;
    int e = (int)(t >> 4);
    if (e >= nEdges) return;
    int g = ((int)t & 15) * 4;
    int src = ei[e];
    int dst = ei[nEdges + e];
    const float4 v = *(const float4*)&feat[(long long)src * 64 + g];
    float* p = &agg[(long long)dst * 64 + g];
    atomicAdd(p + 0, v.x);
    atomicAdd(p + 1, v.y);
    atomicAdd(p + 2, v.z);
    atomicAdd(p + 3, v.w);
}

// Same for feature dim 16 (4 threads per edge).
__global__ void __launch_bounds__(256) gin_scatter16(
    const float* __restrict__ feat, const int* __restrict__ ei,
    float* __restrict__ agg, int nEdges)
{
    long long t = (long long)blockIdx.x * 256 + threadIdx.x;
    int e = (int)(t >> 2);
    if (e >= nEdges) return;
    int g = ((int)t & 3) * 4;
    int src = ei[e];
    int dst = ei[nEdges + e];
    const float4 v = *(const float4*)&feat[(long long)src * 16 + g];
    float* p = &agg[(long long)dst * 16 + g];
    atomicAdd(p + 0, v.x);
    atomicAdd(p + 1, v.y);
    atomicAdd(p + 2, v.z);
    atomicAdd(p + 3, v.w);
}

// ---------------------------------------------------------------------------
// MLP1: h1 = relu(relu(agg1 @ W1 + b1) @ W2 + b2); also seeds agg2 = h1.
// One wave per 16-node tile; 8 waves (256 threads) per block.
// ---------------------------------------------------------------------------
__global__ void __launch_bounds__(256) gin_mlp1(
    const float* __restrict__ agg1,   // [N,64]
    const float* __restrict__ W1, const float* __restrict__ b1,   // [64,16],[16]
    const float* __restrict__ W2, const float* __restrict__ b2,   // [16,16],[16]
    float* __restrict__ h1, float* __restrict__ agg2,             // [N,16] each
    int nNodes)
{
    __shared__ float lds[8][16 * 16];
    const int lane = threadIdx.x & 31;
    const int wave = threadIdx.x >> 5;
    const int tile = blockIdx.x * 8 + wave;
    const int kg   = lane >> 4;        // lane group 0/1 -> K offset +0 / +2
    const int n    = lane & 15;        // column (B/C) == row M (A layout)
    const int row  = tile * 16 + n;    // node index this lane loads for A
    const int crow = row < nNodes ? row : nNodes - 1;  // clamped: always valid
    float* H = lds[wave];

    // ---- layer 1: [16x64] @ [64x16], 16 WMMA steps of K=4 ----
    const float* arow = agg1 + (long long)crow * 64;
    v8f c = {};
    #pragma unroll
    for (int s = 0; s < 16; ++s) {
        int k = s * 4 + kg * 2;
        v2f a = *(const v2f*)(arow + k);          // global_load_b64
        v2f b;
        b.x = W1[k * 16 + n];
        b.y = W1[(k + 1) * 16 + n];
        c = wmma_f32_k4(a, b, c);
    }
    {   // bias + ReLU (C layout: lane holds column n, rows kg*8 + v)
        float bias = b1[n];
        #pragma unroll
        for (int v = 0; v < 8; ++v) c[v] = fmaxf(c[v] + bias, 0.0f);
    }

    // ---- C-layout -> A-layout via LDS ----
    #pragma unroll
    for (int v = 0; v < 8; ++v) H[(kg * 8 + v) * 16 + n] = c[v];
    asm volatile("s_wait_dscnt 0" ::: "memory");

    // ---- layer 2: [16x16] @ [16x16], 4 WMMA steps ----
    v8f c2 = {};
    #pragma unroll
    for (int s = 0; s < 4; ++s) {
        int k = s * 4 + kg * 2;
        v2f a = *(const v2f*)(H + n * 16 + k);    // ds_load_b64 (A row M = n)
        v2f b;
        b.x = W2[k * 16 + n];
        b.y = W2[(k + 1) * 16 + n];
        c2 = wmma_f32_k4(a, b, c2);
    }
    {
        float bias = b2[n];
        #pragma unroll
        for (int v = 0; v < 8; ++v) c2[v] = fmaxf(c2[v] + bias, 0.0f);
    }

    // ---- writeback via LDS so global stores are float4 row-major ----
    #pragma unroll
    for (int v = 0; v < 8; ++v) H[(kg * 8 + v) * 16 + n] = c2[v];
    asm volatile("s_wait_dscnt 0" ::: "memory");

    int m  = lane >> 1;            // row 0..15
    int hh = (lane & 1) * 8;       // half-row offset 0/8
    int grow = tile * 16 + m;
    if (grow < nNodes) {
        float4 p0 = *(const float4*)&H[m * 16 + hh];
        float4 p1 = *(const float4*)&H[m * 16 + hh + 4];
        long long base = (long long)grow * 16 + hh;
        *(float4*)&h1[base]       = p0;
        *(float4*)&h1[base + 4]   = p1;
        *(float4*)&agg2[base]     = p0;   // self term of second aggregate
        *(float4*)&agg2[base + 4] = p1;
    }
}

// ---------------------------------------------------------------------------
// MLP2: out = log_softmax(relu(agg2 @ W3 + b3) @ W4 + b4)
// ---------------------------------------------------------------------------
__global__ void __launch_bounds__(256) gin_mlp2(
    const float* __restrict__ agg2,   // [N,16]
    const float* __restrict__ W3, const float* __restrict__ b3,   // [16,16],[16]
    const float* __restrict__ W4, const float* __restrict__ b4,   // [16,7],[7]
    float* __restrict__ out,                                       // [N,7]
    int nNodes)
{
    __shared__ float lds[8][16 * 16];
    const int lane = threadIdx.x & 31;
    const int wave = threadIdx.x >> 5;
    const int tile = blockIdx.x * 8 + wave;
    const int kg   = lane >> 4;
    const int n    = lane & 15;
    const int row  = tile * 16 + n;
    const int crow = row < nNodes ? row : nNodes - 1;
    float* H = lds[wave];

    // ---- layer 3: [16x16] @ [16x16], 4 WMMA steps ----
    const float* arow = agg2 + (long long)crow * 16;
    v8f c = {};
    #pragma unroll
    for (int s = 0; s < 4; ++s) {
        int k = s * 4 + kg * 2;
        v2f a = *(const v2f*)(arow + k);          // global_load_b64
        v2f b;
        b.x = W3[k * 16 + n];
        b.y = W3[(k + 1) * 16 + n];
        c = wmma_f32_k4(a, b, c);
    }
    {
        float bias = b3[n];
        #pragma unroll
        for (int v = 0; v < 8; ++v) c[v] = fmaxf(c[v] + bias, 0.0f);
    }

    #pragma unroll
    for (int v = 0; v < 8; ++v) H[(kg * 8 + v) * 16 + n] = c[v];
    asm volatile("s_wait_dscnt 0" ::: "memory");

    // ---- classifier (16 -> 7) + log_softmax, one row per lane 0..15 ----
    if (lane < 16) {
        int grow = tile * 16 + lane;
        if (grow < nNodes) {
            float h[16];
            #pragma unroll
            for (int k = 0; k < 16; ++k) h[k] = H[lane * 16 + k];
            float logit[7];
            #pragma unroll
            for (int j = 0; j < 7; ++j) {
                float s = b4[j];
                #pragma unroll
                for (int k = 0; k < 16; ++k) s += h[k] * W4[k * 7 + j];
                logit[j] = s;
            }
            float mx = logit[0];
            #pragma unroll
            for (int j = 1; j < 7; ++j) mx = fmaxf(mx, logit[j]);
            float se = 0.0f;
            #pragma unroll
            for (int j = 0; j < 7; ++j) se += expf(logit[j] - mx);
            float lse = mx + logf(se);
            #pragma unroll
            for (int j = 0; j < 7; ++j) out[(long long)grow * 7 + j] = logit[j] - lse;
        }
    }
}

// ---------------------------------------------------------------------------
extern "C" void kernel_launch(void* const* d_in, const int* in_sizes, int n_in,
                              void* d_out, int out_size, void* d_ws, size_t ws_size,
                              hipStream_t stream) {
    const float* x  = (const float*)d_in[0];   // [N,64]
    const int*   ei = (const int*)d_in[1];     // [2,E] flattened
    const float* W1 = (const float*)d_in[2];
    const float* b1 = (const float*)d_in[3];
    const float* W2 = (const float*)d_in[4];
    const float* b2 = (const float*)d_in[5];
    const float* W3 = (const float*)d_in[6];
    const float* b3 = (const float*)d_in[7];
    const float* W4 = (const float*)d_in[8];
    const float* b4 = (const float*)d_in[9];
    float* out = (float*)d_out;

    const int N = in_sizes[0] / 64;
    const int E = in_sizes[1] / 2;

    // workspace layout
    float* agg1 = (float*)d_ws;                        // N*64
    float* h1   = agg1 + (size_t)N * 64;               // N*16
    float* agg2 = h1   + (size_t)N * 16;               // N*16

    // agg1 = x (self term), then += scatter of x[src]
    hipMemcpyAsync(agg1, x, (size_t)N * 64 * sizeof(float),
                   hipMemcpyDeviceToDevice, stream);
    {
        long long work = (long long)E * 16;
        int blocks = (int)((work + 255) / 256);
        gin_scatter64<<<blocks, 256, 0, stream>>>(x, ei, agg1, E);
    }

    int tiles  = (N + 15) / 16;
    int blocks = (tiles + 7) / 8;
    gin_mlp1<<<blocks, 256, 0, stream>>>(agg1, W1, b1, W2, b2, h1, agg2, N);

    {
        long long work = (long long)E * 4;
        int sblocks = (int)((work + 255) / 256);
        gin_scatter16<<<sblocks, 256, 0, stream>>>(h1, ei, agg2, E);
    }

    gin_mlp2<<<blocks, 256, 0, stream>>>(agg2, W3, b3, W4, b4, out, N);
}